// WeightedLogits_54176717472280
// MI455X (gfx1250) — compile-verified
//
#include <hip/hip_runtime.h>
#include <cstdint>
#include <math.h>

// Problem constants (from reference): B=1024 rows, K = 1024 + 65536 = 66560 cols, f32.
#define KCOLS 66560
#define BROWS 1024
#define TPB   1024                      // 32 wave32 waves per workgroup
#define VEC   4                         // floats per lane per async b128
#define CHUNK (TPB * VEC)               // 4096 floats / chunk / block
#define NCH   ((KCOLS + CHUNK - 1) / CHUNK)   // 17 chunks (last partial)

// ---- scratch layout appended after the row data in dynamic LDS ----
// red[0..31]   : per-wave partial S      (pass 1)  / partial sum2 (pass 2)
// red[32..63]  : per-wave partial M
// red[64]      : l_ii   red[65] : exp(l_ii)
// red[66]      : 1/m    red[67] : rowmax
// red[68..71]  : 16-byte dump slot for clamped tail async loads
#define RED_WORDS 80

// Wait until ASYNCcnt <= n. Literal immediates per case keep the assembler happy
// regardless of unrolling; with #pragma unroll each call folds to one case.
__device__ __forceinline__ void async_wait_le(int n) {
#define WAITC(N) case N: asm volatile("s_wait_asynccnt " #N ::: "memory"); break;
    switch (n) {
        WAITC(0) WAITC(1) WAITC(2) WAITC(3) WAITC(4) WAITC(5) WAITC(6) WAITC(7)
        WAITC(8) WAITC(9) WAITC(10) WAITC(11) WAITC(12) WAITC(13) WAITC(14)
        WAITC(15) WAITC(16)
        default: asm volatile("s_wait_asynccnt 0" ::: "memory"); break;
    }
#undef WAITC
}

// Async copy 16B/lane from global to LDS, non-temporal (stream-once data).
__device__ __forceinline__ void async_load_b128(uint32_t lds_byte_off,
                                                const float* gaddr) {
    unsigned long long ga = (unsigned long long)(uintptr_t)gaddr;
    asm volatile("global_load_async_to_lds_b128 %0, %1, off th:TH_LOAD_NT"
                 :: "v"(lds_byte_off), "v"(ga)
                 : "memory");
}

__device__ __forceinline__ float wave_sum(float v) {
    #pragma unroll
    for (int off = 16; off > 0; off >>= 1) v += __shfl_xor(v, off, 32);
    return v;
}
__device__ __forceinline__ float wave_max(float v) {
    #pragma unroll
    for (int off = 16; off > 0; off >>= 1) v = fmaxf(v, __shfl_xor(v, off, 32));
    return v;
}

__global__ __launch_bounds__(TPB, 1)
void row_loss_kernel(const float* __restrict__ logits, float* __restrict__ row_loss) {
    extern __shared__ float smem[];           // [KCOLS] row data + RED_WORDS scratch
    float* red = smem + KCOLS;
    const int  row  = blockIdx.x;
    const int  tid  = threadIdx.x;
    const int  lane = tid & 31;
    const int  wid  = tid >> 5;
    const float* grow = logits + (size_t)row * KCOLS;
    const uint32_t lds_base = (uint32_t)(uintptr_t)smem;   // LDS offset = addr[31:0]
    const uint32_t dump_off = (uint32_t)(uintptr_t)(red + 68);

    // ---- issue the whole row as async global->LDS traffic (17 instrs/wave) ----
    #pragma unroll
    for (int c = 0; c < NCH; ++c) {
        int idx = c * CHUNK + tid * VEC;
        if (idx <= KCOLS - VEC) {
            async_load_b128(lds_base + (uint32_t)idx * 4u, grow + idx);
        } else {
            // keep EXEC full + per-wave issue count uniform: duplicate-load the
            // row tail into a scratch dump slot (same data from all lanes).
            async_load_b128(dump_off, grow + (KCOLS - VEC));
        }
    }

    // ---- pass 1: S = sum(exp l), M = max off-diag (l*exp l); store t=l*exp l ----
    float S = 0.0f, M = -3.4e38f;
    #pragma unroll
    for (int c = 0; c < NCH; ++c) {
        async_wait_le(NCH - 1 - c);           // chunk c resident; later chunks in flight
        const int base = c * CHUNK + tid * VEC;
        if (base <= KCOLS - VEC) {
            float4 l4 = *reinterpret_cast<const float4*>(smem + base);
            float l[VEC] = {l4.x, l4.y, l4.z, l4.w};
            float t[VEC];
            #pragma unroll
            for (int k = 0; k < VEC; ++k) {
                float e  = __expf(l[k]);
                float tt = l[k] * e;
                S += e;
                t[k] = tt;
                if (base + k == row) { red[64] = l[k]; red[65] = e; }
                else                 { M = fmaxf(M, tt); }
            }
            *reinterpret_cast<float4*>(smem + base) = make_float4(t[0], t[1], t[2], t[3]);
        }
    }

    S = wave_sum(S);
    M = wave_max(M);
    if (lane == 0) { red[wid] = S; red[32 + wid] = M; }
    __syncthreads();
    if (tid < 32) {
        float s = wave_sum(red[tid]);
        float m = wave_max(red[32 + tid]);
        if (tid == 0) {
            float l_ii = red[64], e_ii = red[65];
            float nm   = (s - e_ii) / (float)(KCOLS - 1);   // neg_mean
            float inv  = 1.0f / nm;
            float rmax = fmaxf(m * inv, l_ii);
            red[66] = inv;
            red[67] = rmax;
            smem[row] = l_ii * nm;   // uniformize diagonal: (l_ii*nm)*inv == l_ii
        }
    }
    __syncthreads();

    // ---- pass 2: logsumexp over u = t/m (all from LDS, zero HBM traffic) ----
    const float inv  = red[66];
    const float rmax = red[67];
    float s2 = 0.0f;
    #pragma unroll
    for (int c = 0; c < NCH; ++c) {
        const int base = c * CHUNK + tid * VEC;
        if (base <= KCOLS - VEC) {
            float4 t4 = *reinterpret_cast<const float4*>(smem + base);
            s2 += __expf(t4.x * inv - rmax) + __expf(t4.y * inv - rmax)
                + __expf(t4.z * inv - rmax) + __expf(t4.w * inv - rmax);
        }
    }
    s2 = wave_sum(s2);
    __syncthreads();                      // reduction-1 reads of red[0..31] are done
    if (lane == 0) red[wid] = s2;
    __syncthreads();
    if (tid < 32) {
        float v = wave_sum(red[tid]);
        if (tid == 0) {
            float lse = rmax + __logf(v);
            row_loss[row] = lse - red[64];   // -logp[i,i]
        }
    }
}

__global__ __launch_bounds__(BROWS)
void final_reduce_kernel(const float* __restrict__ row_loss, float* __restrict__ out) {
    __shared__ float red[32];
    const int tid = threadIdx.x;
    float v = wave_sum(row_loss[tid]);
    if ((tid & 31) == 0) red[tid >> 5] = v;
    __syncthreads();
    if (tid < 32) {
        float w = wave_sum(red[tid]);
        if (tid == 0) out[0] = w * (1.0f / (float)BROWS);
    }
}

extern "C" void kernel_launch(void* const* d_in, const int* in_sizes, int n_in,
                              void* d_out, int out_size, void* d_ws, size_t ws_size,
                              hipStream_t stream) {
    (void)in_sizes; (void)n_in; (void)out_size; (void)ws_size;
    const float* logits = (const float*)d_in[0];   // [B, K] f32
    float* out  = (float*)d_out;                   // scalar f32
    float* rows = (float*)d_ws;                    // B per-row losses (scratch)

    const size_t shmem = (size_t)(KCOLS + RED_WORDS) * sizeof(float);  // ~266.6 KB < 320 KB WGP LDS
    row_loss_kernel<<<BROWS, TPB, shmem, stream>>>(logits, rows);
    final_reduce_kernel<<<1, BROWS, 0, stream>>>(rows, out);
}